// GaussianRenderer_1743756722798
// MI455X (gfx1250) — compile-verified
//
#include <hip/hip_runtime.h>
#include <hip/hip_bf16.h>

// ---------------------------------------------------------------------------
// Gaussian splat renderer for gfx1250 (MI455X), wave32.
//   prep_kernel:   1 block / camera, 1024 threads: project + bitonic sort by z,
//                  gather into a PAIR-PACKED workspace layout (80B per pair of
//                  sorted gaussians) so the render loop's ds_load_b128s land
//                  directly in v_pk_f32 operand order (no register shuffles).
//   render_kernel: grid (392, 2), 256 threads (8 waves). Each wave renders a
//                  16-pixel tile. 40KB of params staged into LDS with
//                  GLOBAL_LOAD_ASYNC_TO_LDS_B128 (ASYNCcnt path). Compositing
//                  accumulates on the matrix pipe via V_WMMA_F32_16X16X4_F32
//                  (A = 16x4 prefix-weighted alphas, B = 4x16 [r g b 1 0...]),
//                  so D[:,0..2] = img pre-background, D[:,3] = 1 - T_final.
// ---------------------------------------------------------------------------

#define HW 224
#define NPIX (HW * HW)
#define NG 1024
#define NCAM 2
#define ZNEAR 0.01f
#define ZFAR 100.0f
#define EPS2D 0.3f
#define ALPHA_MIN (1.0f / 255.0f)
#define ALPHA_MAX 0.999f
#define PAIR_FLOATS 20                   // 80B per gaussian pair (16B aligned)
#define CAM_STRIDE (NG / 2 * PAIR_FLOATS)  // 10240 floats = 40KB per camera
#define IMG_TOTAL (NCAM * NPIX * 3)

typedef __attribute__((ext_vector_type(2))) float v2f;
typedef __attribute__((ext_vector_type(8))) float v8f;

// ---------------------------------------------------------------------------
// Kernel 1: per-gaussian projection + bitonic argsort by z + sorted gather.
// ---------------------------------------------------------------------------
__global__ __launch_bounds__(NG) void prep_kernel(
    const float* __restrict__ w2cs, const float* __restrict__ Ks,
    const float* __restrict__ xyz, const float* __restrict__ rgb,
    const float* __restrict__ opacity, const float* __restrict__ scale,
    const float* __restrict__ rotation, float* __restrict__ ws) {
  const int cam = blockIdx.x;
  const int g = threadIdx.x;

  __shared__ float s_key[NG];
  __shared__ int s_idx[NG];
  __shared__ float s_par[9][NG];  // u v ia ib id o r g b (unsorted)

  // --- quaternion -> rotation, cov3d = (R*S)(R*S)^T -------------------------
  float qw = rotation[g * 4 + 0], qx = rotation[g * 4 + 1];
  float qy = rotation[g * 4 + 2], qz = rotation[g * 4 + 3];
  float qn = rsqrtf(qw * qw + qx * qx + qy * qy + qz * qz);
  qw *= qn; qx *= qn; qy *= qn; qz *= qn;
  float R00 = 1.f - 2.f * (qy * qy + qz * qz), R01 = 2.f * (qx * qy - qw * qz), R02 = 2.f * (qx * qz + qw * qy);
  float R10 = 2.f * (qx * qy + qw * qz), R11 = 1.f - 2.f * (qx * qx + qz * qz), R12 = 2.f * (qy * qz - qw * qx);
  float R20 = 2.f * (qx * qz - qw * qy), R21 = 2.f * (qy * qz + qw * qx), R22 = 1.f - 2.f * (qx * qx + qy * qy);
  float sx = scale[g * 3 + 0], sy = scale[g * 3 + 1], sz = scale[g * 3 + 2];
  float M00 = R00 * sx, M01 = R01 * sy, M02 = R02 * sz;
  float M10 = R10 * sx, M11 = R11 * sy, M12 = R12 * sz;
  float M20 = R20 * sx, M21 = R21 * sy, M22 = R22 * sz;
  float c00 = M00 * M00 + M01 * M01 + M02 * M02;
  float c01 = M00 * M10 + M01 * M11 + M02 * M12;
  float c02 = M00 * M20 + M01 * M21 + M02 * M22;
  float c11 = M10 * M10 + M11 * M11 + M12 * M12;
  float c12 = M10 * M20 + M11 * M21 + M12 * M22;
  float c22 = M20 * M20 + M21 * M21 + M22 * M22;

  // --- camera transform -----------------------------------------------------
  const float* V = w2cs + cam * 16;
  float x = xyz[g * 3 + 0], y = xyz[g * 3 + 1], z3 = xyz[g * 3 + 2];
  float pcx = V[0] * x + V[1] * y + V[2] * z3 + V[3];
  float pcy = V[4] * x + V[5] * y + V[6] * z3 + V[7];
  float pcz = V[8] * x + V[9] * y + V[10] * z3 + V[11];
  const float* K = Ks + cam * 9;
  float fx = K[0], cx = K[2], fy = K[4], cy = K[5];
  float rz = 1.0f / pcz;
  float limx = 1.3f * (0.5f * HW / fx), limy = 1.3f * (0.5f * HW / fy);
  float tx = pcz * fminf(fmaxf(pcx * rz, -limx), limx);
  float ty = pcz * fminf(fmaxf(pcy * rz, -limy), limy);

  // cov_cam = Rv * cov3d * Rv^T (Rv = view rotation rows)
  float v0x = V[0], v0y = V[1], v0z = V[2];
  float v1x = V[4], v1y = V[5], v1z = V[6];
  float v2x = V[8], v2y = V[9], v2z = V[10];
  float t0x = v0x * c00 + v0y * c01 + v0z * c02;
  float t0y = v0x * c01 + v0y * c11 + v0z * c12;
  float t0z = v0x * c02 + v0y * c12 + v0z * c22;
  float t1x = v1x * c00 + v1y * c01 + v1z * c02;
  float t1y = v1x * c01 + v1y * c11 + v1z * c12;
  float t1z = v1x * c02 + v1y * c12 + v1z * c22;
  float t2x = v2x * c00 + v2y * c01 + v2z * c02;
  float t2y = v2x * c01 + v2y * c11 + v2z * c12;
  float t2z = v2x * c02 + v2y * c12 + v2z * c22;
  float cc00 = t0x * v0x + t0y * v0y + t0z * v0z;
  float cc01 = t0x * v1x + t0y * v1y + t0z * v1z;
  float cc02 = t0x * v2x + t0y * v2y + t0z * v2z;
  float cc11 = t1x * v1x + t1y * v1y + t1z * v1z;
  float cc12 = t1x * v2x + t1y * v2y + t1z * v2z;
  float cc22 = t2x * v2x + t2y * v2y + t2z * v2z;

  // J (2x3), cov2d = J cc J^T
  float j00 = fx * rz, j02 = -fx * tx * rz * rz;
  float j11 = fy * rz, j12 = -fy * ty * rz * rz;
  float Jc0x = j00 * cc00 + j02 * cc02;
  float Jc0y = j00 * cc01 + j02 * cc12;
  float Jc0z = j00 * cc02 + j02 * cc22;
  float Jc1y = j11 * cc11 + j12 * cc12;
  float Jc1z = j11 * cc12 + j12 * cc22;
  float a2 = Jc0x * j00 + Jc0z * j02 + EPS2D;
  float b2 = Jc0y * j11 + Jc0z * j12;
  float d2 = Jc1y * j11 + Jc1z * j12 + EPS2D;
  float det = a2 * d2 - b2 * b2;
  bool valid = (pcz > ZNEAR) && (pcz < ZFAR) && (det > 0.0f);
  float idet = 1.0f / det;
  float ia = valid ? d2 * idet : 0.0f;
  float id_ = valid ? a2 * idet : 0.0f;
  float ib = valid ? -b2 * idet : 0.0f;
  float u = valid ? fx * pcx * rz + cx : 0.0f;
  float v = valid ? fy * pcy * rz + cy : 0.0f;
  float o = valid ? opacity[g] : 0.0f;

  s_par[0][g] = u;  s_par[1][g] = v;  s_par[2][g] = ia;
  s_par[3][g] = ib; s_par[4][g] = id_; s_par[5][g] = o;
  s_par[6][g] = rgb[g * 3 + 0]; s_par[7][g] = rgb[g * 3 + 1]; s_par[8][g] = rgb[g * 3 + 2];
  s_key[g] = pcz;   // reference sorts raw z (invalids contribute 0 anyway)
  s_idx[g] = g;

  // --- bitonic sort (key=z ascending, payload=index) ------------------------
  for (int kk = 2; kk <= NG; kk <<= 1) {
    for (int j = kk >> 1; j > 0; j >>= 1) {
      __syncthreads();
      int ixj = g ^ j;
      if (ixj > g) {
        float ka = s_key[g], kb = s_key[ixj];
        bool ascending = ((g & kk) == 0);
        if ((ka > kb) == ascending) {
          s_key[g] = kb; s_key[ixj] = ka;
          int t = s_idx[g]; s_idx[g] = s_idx[ixj]; s_idx[ixj] = t;
        }
      }
    }
  }
  __syncthreads();

  // --- gather in sorted order, write pair-packed workspace ------------------
  // Pair layout (80B): {u0,u1,v0,v1 | ia0,ia1,ib0,ib1 | id0,id1,o0,o1 |
  //                     r0,r1,g0,g1 | b0,b1,pad,pad}
  int s = s_idx[g];
  float* pb = ws + cam * CAM_STRIDE + (g >> 1) * PAIR_FLOATS;
  int slot = g & 1;
  pb[0 + slot] = s_par[0][s];   // u
  pb[2 + slot] = s_par[1][s];   // v
  pb[4 + slot] = s_par[2][s];   // ia
  pb[6 + slot] = s_par[3][s];   // ib
  pb[8 + slot] = s_par[4][s];   // id
  pb[10 + slot] = s_par[5][s];  // o
  pb[12 + slot] = s_par[6][s];  // r
  pb[14 + slot] = s_par[7][s];  // g
  pb[16 + slot] = s_par[8][s];  // b
}

// ---------------------------------------------------------------------------
// Kernel 2: compositing. Per wave: 16 pixels, WMMA f32 16x16x4 accumulation.
// ---------------------------------------------------------------------------
__global__ __launch_bounds__(256) void render_kernel(const float* __restrict__ ws,
                                                     float* __restrict__ out) {
  __shared__ __align__(16) float s_pairs[CAM_STRIDE];  // 40KB, pair-packed

  const int cam = blockIdx.y;
  const float* base = ws + cam * CAM_STRIDE;

  // --- stage 40KB of sorted gaussian params via async global->LDS DMA -------
  // GLOBAL_LOAD_ASYNC_TO_LDS_B128: VDST = LDS byte address (VGPR), VADDR =
  // 64-bit global address; tracked by ASYNCcnt (CDNA5 ISA). Flat copy:
  // 40960B = 2560 x 16B chunks, 10 per thread.
  {
    unsigned lbase = (unsigned)(size_t)&s_pairs[0];
    for (int i = threadIdx.x; i < CAM_STRIDE / 4; i += 256) {
      unsigned l = lbase + 16u * i;
      unsigned long long ga = (unsigned long long)base + 16ull * i;
      asm volatile("global_load_async_to_lds_b128 %0, %1, off" ::"v"(l), "v"(ga) : "memory");
    }
    asm volatile("s_wait_asynccnt 0x0" ::: "memory");
  }
  __syncthreads();

  const int lane = threadIdx.x & 31;
  const int half = lane >> 4;      // which K-pair of the 4-wide chunk
  const int col = lane & 15;       // N column of B / pixel M row of A
  const int wave = threadIdx.x >> 5;
  const int tile = blockIdx.x * 128 + wave * 16;  // 16 pixels per wave

  const int mypix = tile + col;    // pixel this lane evaluates alphas for
  const float px = (float)(mypix % HW) + 0.5f;
  const float py = (float)(mypix / HW) + 0.5f;

  // Branchless B-column selectors (hoisted): B column is [r g b 1 0...0].
  const float selr = (col == 0) ? 1.0f : 0.0f;
  const float selg = (col == 1) ? 1.0f : 0.0f;
  const float selb = (col == 2) ? 1.0f : 0.0f;
  const float sel1 = (col == 3) ? 1.0f : 0.0f;

  float T = 1.0f;                  // transmittance for pixel `col` (both halves)
  v8f acc = {};                    // D: 16 pixels x [r g b (1-Tf) 0...]

  // Each lane walks its half's pair: +2 pairs (160B) per 4-gaussian chunk.
  const float* pb = s_pairs + half * PAIR_FLOATS;

#pragma unroll 4
  for (int kb = 0; kb < NG; kb += 4, pb += 2 * PAIR_FLOATS) {
    float u0 = pb[0], u1 = pb[1], v0 = pb[2], v1 = pb[3];
    float ia0 = pb[4], ia1 = pb[5], ib0 = pb[6], ib1 = pb[7];
    float id0 = pb[8], id1 = pb[9], o0 = pb[10], o1 = pb[11];
    float r0 = pb[12], r1 = pb[13], g0 = pb[14], g1 = pb[15];
    float b0 = pb[16], b1 = pb[17];

    float dx0 = u0 - px, dy0 = v0 - py;
    float dx1 = u1 - px, dy1 = v1 - py;
    float s0 = 0.5f * (ia0 * dx0 * dx0 + id0 * dy0 * dy0) + ib0 * dx0 * dy0;
    float s1 = 0.5f * (ia1 * dx1 * dx1 + id1 * dy1 * dy1) + ib1 * dx1 * dy1;
    float al0 = fminf(o0 * __expf(-s0), ALPHA_MAX);
    float al1 = fminf(o1 * __expf(-s1), ALPHA_MAX);
    float a0 = (s0 >= 0.0f && al0 > ALPHA_MIN) ? al0 : 0.0f;
    float a1 = (s1 >= 0.0f && al1 > ALPHA_MIN) ? al1 : 0.0f;

    float na0 = 1.0f - a0;
    float pg = na0 * (1.0f - a1);
    float og = __shfl_xor(pg, 16);          // complementary half's product
    float pre = half ? T * og : T;          // exact prefix transmittance

    v2f A;                                   // A[M=col][K = 2*half + {0,1}]
    A.x = a0 * pre;
    A.y = a1 * pre * na0;
    T *= pg * og;                            // advance past all 4 gaussians

    v2f B;                                   // B[K = 2*half + {0,1}][N=col]
    B.x = fmaf(selr, r0, fmaf(selg, g0, fmaf(selb, b0, sel1)));
    B.y = fmaf(selr, r1, fmaf(selg, g1, fmaf(selb, b1, sel1)));

    acc = __builtin_amdgcn_wmma_f32_16x16x4_f32(
        /*neg_a=*/false, A, /*neg_b=*/false, B,
        /*c_mod=*/(short)0, acc, /*reuse_a=*/false, /*reuse_b=*/false);
  }

  // D layout: VGPR r, lanes 0-15 -> (M=r, N=lane); lanes 16-31 -> (M=r+8, N=lane-16)
#pragma unroll
  for (int r = 0; r < 8; ++r) {
    float vv = acc[r];
    float ap = __shfl(vv, (half << 4) + 3);  // channel-3 value = 1 - Tf for pixel M
    int pixel = tile + r + 8 * half;
    int pyi = pixel / HW, pxi = pixel % HW;
    if (col < 3) {
      // img = accumulated color + Tf * bg  (bg = 1)
      out[((cam * HW + pyi) * HW + pxi) * 3 + col] = vv + (1.0f - ap);
    } else if (col == 3) {
      out[IMG_TOTAL + (cam * HW + pyi) * HW + pxi] = ap;
    }
  }
}

extern "C" void kernel_launch(void* const* d_in, const int* in_sizes, int n_in,
                              void* d_out, int out_size, void* d_ws, size_t ws_size,
                              hipStream_t stream) {
  const float* w2cs = (const float*)d_in[0];
  const float* Ks = (const float*)d_in[1];
  const float* xyz = (const float*)d_in[2];
  const float* rgb = (const float*)d_in[3];
  const float* opacity = (const float*)d_in[4];
  const float* scale = (const float*)d_in[5];
  const float* rotation = (const float*)d_in[6];
  float* out = (float*)d_out;
  float* ws = (float*)d_ws;

  prep_kernel<<<dim3(NCAM), dim3(NG), 0, stream>>>(w2cs, Ks, xyz, rgb, opacity,
                                                   scale, rotation, ws);
  render_kernel<<<dim3(NPIX / 128, NCAM), dim3(256), 0, stream>>>(ws, out);
}